// controller_1340029796839
// MI455X (gfx1250) — compile-verified
//
#include <hip/hip_runtime.h>
#include <cstdint>

#define LSTM_H 3
#define GATES  12   // 4*H

// ---------------------------------------------------------------------------
// gfx1250 async global->LDS staging (ASYNCcnt-tracked DMA path).
// Signature probe-confirmed:
//   void __builtin_amdgcn_global_load_async_to_lds_b128(
//        v4i addrspace(1)* gsrc, v4i addrspace(3)* ldst, imm offset, imm cpol)
// For a generic "shared" pointer the low 32 bits ARE the LDS offset
// (ISA 10.2: the LDS address is the low 32 address bits), so the integer casts are exact.
// ---------------------------------------------------------------------------
typedef int v4i __attribute__((vector_size(4 * sizeof(int))));
typedef __attribute__((address_space(1))) v4i* gas_v4i_ptr;
typedef __attribute__((address_space(3))) v4i* las_v4i_ptr;

#if __has_builtin(__builtin_amdgcn_global_load_async_to_lds_b128)
#define HAVE_ASYNC_BUILTIN 1
#endif

__device__ __forceinline__ void copy16_to_lds(const float* __restrict__ src,
                                              float* dst) {
#if defined(HAVE_ASYNC_BUILTIN)
  __builtin_amdgcn_global_load_async_to_lds_b128(
      (gas_v4i_ptr)(uintptr_t)src,
      (las_v4i_ptr)(uint32_t)(uintptr_t)dst,
      /*offset=*/0, /*cpol=*/0);
#else
  asm volatile("global_load_async_to_lds_b128 %0, %1, off"
               :
               : "v"((uint32_t)(uintptr_t)dst), "v"((uint64_t)(uintptr_t)src)
               : "memory");
#endif
}

__device__ __forceinline__ void wait_async_all() {
#if __has_builtin(__builtin_amdgcn_s_wait_asynccnt)
  __builtin_amdgcn_s_wait_asynccnt(0);
#else
  asm volatile("s_wait_asynccnt 0" ::: "memory");
#endif
}

__device__ __forceinline__ void stage_region(const float* __restrict__ src,
                                             float* dst, int nf, int tid, int T) {
  const int n4 = nf >> 2;
  for (int q = tid; q < n4; q += T)
    copy16_to_lds(src + (q << 2), dst + (q << 2));
  for (int r = (n4 << 2) + tid; r < nf; r += T)   // tail (none for n%4==0)
    dst[r] = src[r];
}

// ---------------------------------------------------------------------------
// Fast activations on the TRANS pipe (gfx1250 v_tanh_f32 when available).
// ---------------------------------------------------------------------------
#if __has_builtin(__builtin_amdgcn_tanhf)
#define TANHF(x) __builtin_amdgcn_tanhf(x)
#define HAVE_TANH 1
#elif __has_builtin(__builtin_amdgcn_tanh_f32)
#define TANHF(x) __builtin_amdgcn_tanh_f32(x)
#define HAVE_TANH 1
#endif

__device__ __forceinline__ float fast_tanh(float x) {
#if defined(HAVE_TANH)
  return TANHF(x);
#else
  const float t = __expf(-2.0f * x);             // tanh = 2/(1+e^-2x) - 1
  return fmaf(2.0f, __builtin_amdgcn_rcpf(1.0f + t), -1.0f);
#endif
}

__device__ __forceinline__ float fast_sigmoid(float x) {
#if defined(HAVE_TANH)
  return fmaf(0.5f, TANHF(0.5f * x), 0.5f);      // sigmoid via v_tanh_f32
#else
  return __builtin_amdgcn_rcpf(1.0f + __expf(-x));
#endif
}

// ---------------------------------------------------------------------------
// One block.
//  Phase 1: all 8 waves DMA 292KB (whole problem) into LDS via async-to-LDS.
//  Phase 2: all 256 threads permute each cell's blocks IN PLACE to a
//           per-lane-contiguous layout and fold bih+bhh (bijective within
//           each cell's own block -> no extra LDS, no races).
//  Phase 3: wave 0 runs the serial chain, software-pipelined: iteration i
//           issues iteration i+1's 6 LDS loads so ds latency hides under the
//           activation chain. Lane k (k<3) owns row k of (h,c); the only
//           cross-lane op on the chain is broadcasting h.
// ---------------------------------------------------------------------------
__global__ void __launch_bounds__(256, 1)
lstm_chain_kernel(const float* __restrict__ state,
                  const float* __restrict__ Wih,
                  const float* __restrict__ Whh,
                  const float* __restrict__ bih,
                  const float* __restrict__ bhh,
                  float* __restrict__ out, int n) {
  extern __shared__ float L[];

  const int nA      = (n + 3) & ~3;          // keep 16B alignment per region
  const int off_s   = 0;
  const int off_wih = nA;                    // [cell][k*4+g4]      after pre-pass
  const int off_whh = off_wih + 12 * n;      // [cell][k*12+g4*3+c] after pre-pass
  const int off_bih = off_whh + 36 * n;      // [cell][k*4+g4] = bih+bhh after pre-pass
  const int off_bhh = off_bih + 12 * n;      // staging input only

  const int tid = threadIdx.x;
  const int T   = blockDim.x;

  // ---- Phase 1: async DMA everything into LDS --------------------------------
  stage_region(state, L + off_s,   n,      tid, T);
  stage_region(Wih,   L + off_wih, 12 * n, tid, T);
  stage_region(Whh,   L + off_whh, 36 * n, tid, T);
  stage_region(bih,   L + off_bih, 12 * n, tid, T);
  stage_region(bhh,   L + off_bhh, 12 * n, tid, T);
  wait_async_all();
  __syncthreads();

  // ---- Phase 2: in-place per-cell permutation (one thread per cell) ----------
  // gate row r = g4*3+k  (g4: 0=i,1=f,2=g,3=o ; k: hidden row)
  for (int cell = tid; cell < n; cell += T) {
    float* pw = &L[off_wih + cell * 12];
    float twi[12];
#pragma unroll
    for (int r = 0; r < 12; ++r) twi[r] = pw[r];
#pragma unroll
    for (int g4 = 0; g4 < 4; ++g4)
#pragma unroll
      for (int k = 0; k < 3; ++k) pw[k * 4 + g4] = twi[g4 * 3 + k];

    float* pb = &L[off_bih + cell * 12];
    const float* ph2 = &L[off_bhh + cell * 12];
    float tb[12];
#pragma unroll
    for (int r = 0; r < 12; ++r) tb[r] = pb[r] + ph2[r];
#pragma unroll
    for (int g4 = 0; g4 < 4; ++g4)
#pragma unroll
      for (int k = 0; k < 3; ++k) pb[k * 4 + g4] = tb[g4 * 3 + k];

    float* pm = &L[off_whh + cell * 36];
    float tw[36];
#pragma unroll
    for (int q = 0; q < 36; ++q) tw[q] = pm[q];
#pragma unroll
    for (int g4 = 0; g4 < 4; ++g4)
#pragma unroll
      for (int k = 0; k < 3; ++k)
#pragma unroll
        for (int c = 0; c < 3; ++c)
          pm[k * 12 + g4 * 3 + c] = tw[(g4 * 3 + k) * 3 + c];
  }
  __syncthreads();

  // ---- Phase 3: serial chain on wave 0, register double-buffered -------------
  if (tid < 32) {
    const int lane = tid;
    const int k = (lane < LSTM_H) ? lane : 0;   // EXEC stays all-ones
    float h0 = 0.0f, h1 = 0.0f, h2 = 0.0f, c = 0.0f;

    // prologue: load step 0
    float  x  = L[off_s];
    float4 wv = *(const float4*)&L[off_wih + k * 4];
    float4 bv = *(const float4*)&L[off_bih + k * 4];
    float4 wa = *(const float4*)&L[off_whh + k * 12 + 0];
    float4 wb = *(const float4*)&L[off_whh + k * 12 + 4];
    float4 wc = *(const float4*)&L[off_whh + k * 12 + 8];

    for (int i = 0; i < n; ++i) {
      // issue next step's loads first: ds latency overlaps this step's chain
      const int j = (i + 1 < n) ? (i + 1) : i;  // last iter: harmless reload
      const float  xn  = L[off_s + j];
      const float4 wvn = *(const float4*)&L[off_wih + j * 12 + k * 4];
      const float4 bvn = *(const float4*)&L[off_bih + j * 12 + k * 4];
      const float4 wan = *(const float4*)&L[off_whh + j * 36 + k * 12 + 0];
      const float4 wbn = *(const float4*)&L[off_whh + j * 36 + k * 12 + 4];
      const float4 wcn = *(const float4*)&L[off_whh + j * 36 + k * 12 + 8];

      // z[g4] = wih*x + (bih+bhh) + Whh_row . h   (components per layout above)
      float zi = fmaf(wv.x, x, bv.x);            // gate i
      zi = fmaf(wa.x, h0, zi); zi = fmaf(wa.y, h1, zi); zi = fmaf(wa.z, h2, zi);
      float zf = fmaf(wv.y, x, bv.y);            // gate f
      zf = fmaf(wa.w, h0, zf); zf = fmaf(wb.x, h1, zf); zf = fmaf(wb.y, h2, zf);
      float zg = fmaf(wv.z, x, bv.z);            // gate g
      zg = fmaf(wb.z, h0, zg); zg = fmaf(wb.w, h1, zg); zg = fmaf(wc.x, h2, zg);
      float zo = fmaf(wv.w, x, bv.w);            // gate o
      zo = fmaf(wc.y, h0, zo); zo = fmaf(wc.z, h1, zo); zo = fmaf(wc.w, h2, zo);

      const float ai = fast_sigmoid(zi);
      const float af = fast_sigmoid(zf);
      const float ag = fast_tanh(zg);
      const float ao = fast_sigmoid(zo);

      c = fmaf(af, c, ai * ag);
      const float h = ao * fast_tanh(c);

      // broadcast new h to every lane (only cross-lane op on the chain)
      h0 = __shfl(h, 0);
      h1 = __shfl(h, 1);
      h2 = __shfl(h, 2);

      // softmax over the 3 h values, fully in-lane from broadcasts
      const float m  = fmaxf(fmaxf(h0, h1), h2);
      const float e0 = __expf(h0 - m);
      const float e1 = __expf(h1 - m);
      const float e2 = __expf(h2 - m);
      const float rs = __builtin_amdgcn_rcpf(e0 + e1 + e2);
      if (lane < LSTM_H) out[i * LSTM_H + lane] = __expf(h - m) * rs;

      // rotate double buffer
      x = xn; wv = wvn; bv = bvn; wa = wan; wb = wbn; wc = wcn;
    }
  }
}

// ---------------------------------------------------------------------------
extern "C" void kernel_launch(void* const* d_in, const int* in_sizes, int n_in,
                              void* d_out, int out_size, void* d_ws, size_t ws_size,
                              hipStream_t stream) {
  (void)n_in; (void)out_size; (void)d_ws; (void)ws_size;
  const float* state = (const float*)d_in[0];
  const float* Wih   = (const float*)d_in[1];
  const float* Whh   = (const float*)d_in[2];
  const float* bih   = (const float*)d_in[3];
  const float* bhh   = (const float*)d_in[4];
  float* out = (float*)d_out;

  const int n  = in_sizes[0];                     // N = 1000
  const int nA = (n + 3) & ~3;
  const size_t ldsBytes = (size_t)(nA + 72 * n) * sizeof(float);  // 292 KB

  lstm_chain_kernel<<<dim3(1), dim3(256), ldsBytes, stream>>>(
      state, Wih, Whh, bih, bhh, out, n);
}